// SokEmbLayer_755914244423
// MI455X (gfx1250) — compile-verified
//
#include <hip/hip_runtime.h>
#include <hip/hip_bf16.h>
#include <stdint.h>

// Problem constants (from the reference)
#define NUM_TABLES 26
#define VOCAB      100000
#define EMB_DIM    128
#define BATCH      8192
#define HOTNESS    10

// Kernel shape
#define WAVES_PER_BLOCK 8
#define ROWS_PER_WAVE   4
#define THREADS        (WAVES_PER_BLOCK * 32)
#define ROW_FLOATS      EMB_DIM              // 128 floats per gathered row
#define ROW_BYTES      (EMB_DIM * 4)         // 512 B per gathered row
#define BUF_FLOATS     (HOTNESS * EMB_DIM)   // one pipeline buffer = 10 rows = 5120 B

typedef float __attribute__((ext_vector_type(4))) f32x4;

// One wave per 4 output rows (table t, batches b0..b0+3). Lane owns 4 consecutive floats of D.
// Gathers go memory -> LDS via CDNA5 async copies (ASYNCcnt), double-buffered so
// row r+1's 10 gathers are in flight while row r is reduced from LDS.
__global__ __launch_bounds__(THREADS) void sok_emb_mean_gather(
    const int* __restrict__ ids,       // [T, B, H] int32
    const float* __restrict__ tables,  // [T, V, D] f32
    float* __restrict__ out)           // [B, T, D] f32
{
  __shared__ __align__(16) float lds_buf[WAVES_PER_BLOCK * 2 * BUF_FLOATS];  // 80 KB

  const int lane = (int)(threadIdx.x & 31u);
  const int wave = __builtin_amdgcn_readfirstlane((int)(threadIdx.x >> 5));
  const int t    = (int)blockIdx.y;
  const int b0   = ((int)blockIdx.x * WAVES_PER_BLOCK + wave) * ROWS_PER_WAVE;

  // Uniform 64-bit table base for the SADDR (GVS) form of the async copy.
  const uint64_t tab_base =
      (uint64_t)(uintptr_t)(tables + (size_t)t * VOCAB * EMB_DIM);
  const int* idr = ids + ((size_t)t * BATCH + (size_t)b0) * HOTNESS;

  // Pre-fetch ALL ids this wave needs (40 consecutive int32 = 160 B) into
  // SGPRs before any asm: the loads precede every "memory"-clobbering asm, so
  // the compiler merges them into wide s_loads with a single s_wait_kmcnt.
  int idv[ROWS_PER_WAVE * HOTNESS];
#pragma unroll
  for (int i = 0; i < ROWS_PER_WAVE * HOTNESS; ++i)
    idv[i] = __builtin_amdgcn_readfirstlane(idr[i]);

  float* mybuf = &lds_buf[(size_t)wave * 2 * BUF_FLOATS];
  // Low 32 bits of a generic pointer into __shared__ == raw LDS byte offset.
  const uint32_t lds0   = (uint32_t)(uintptr_t)mybuf + (uint32_t)(lane * 16);
  const uint32_t lane16 = (uint32_t)(lane * 16);

  auto issue_row = [&](int r, int buf) {
    const uint32_t dst0 = lds0 + (uint32_t)buf * (uint32_t)(BUF_FLOATS * 4);
#pragma unroll
    for (int h = 0; h < HOTNESS; ++h) {
      // 32-bit gather offset: id*512 + lane*16  (max ~51 MB, fits signed i32)
      const uint32_t off = (uint32_t)idv[r * HOTNESS + h] * (uint32_t)ROW_BYTES
                         + lane16;
      const uint32_t dst = dst0 + (uint32_t)(h * ROW_BYTES);
      // CDNA5 async copy, GVS mode: 16 B/lane, memory -> LDS, ASYNCcnt-tracked.
      asm volatile("global_load_async_to_lds_b128 %0, %1, %2"
                   :: "v"(dst), "v"(off), "s"(tab_base)
                   : "memory");
    }
  };

  issue_row(0, 0);

#pragma unroll
  for (int r = 0; r < ROWS_PER_WAVE; ++r) {
    const int cur = r & 1;
    if (r + 1 < ROWS_PER_WAVE) {
      issue_row(r + 1, cur ^ 1);
      // Async completions are in-order: <=10 outstanding => row r's 10 landed.
      asm volatile("s_wait_asynccnt 10" ::: "memory");
    } else {
      asm volatile("s_wait_asynccnt 0" ::: "memory");
    }

    const float* bp = mybuf + (size_t)cur * BUF_FLOATS + (size_t)(lane * 4);
    f32x4 acc = {0.f, 0.f, 0.f, 0.f};
#pragma unroll
    for (int h = 0; h < HOTNESS; ++h) {
      const f32x4 v = *(const f32x4*)(bp + h * ROW_FLOATS);  // ds_load_b128
      acc += v;
    }
    acc *= (1.0f / (float)HOTNESS);  // mean combiner

    float* dst = out + (((size_t)(b0 + r)) * NUM_TABLES + (size_t)t) * EMB_DIM
                     + (size_t)(lane * 4);
    // Write-once output: non-temporal so the 109 MB stream doesn't evict
    // table rows from the 192 MB L2 (per-table reuse ~1.46x).
    __builtin_nontemporal_store(acc, (f32x4*)dst);
  }
}

extern "C" void kernel_launch(void* const* d_in, const int* in_sizes, int n_in,
                              void* d_out, int out_size, void* d_ws, size_t ws_size,
                              hipStream_t stream) {
  (void)in_sizes; (void)n_in; (void)out_size; (void)d_ws; (void)ws_size;
  const int*   ids    = (const int*)d_in[0];   // [T,B,H] int32
  const float* tables = (const float*)d_in[1]; // [T,V,D] f32
  float*       out    = (float*)d_out;         // [B,T,D] f32

  // x = batch chunks (fast-varying), y = table (slow) -> same-table blocks are
  // adjacent in launch order, keeping each table's ~29 MB hot set in L2.
  dim3 grid(BATCH / (WAVES_PER_BLOCK * ROWS_PER_WAVE), NUM_TABLES, 1);
  sok_emb_mean_gather<<<grid, THREADS, 0, stream>>>(ids, tables, out);
}